// GAT_24481313587814
// MI455X (gfx1250) — compile-verified
//
#include <hip/hip_runtime.h>
#include <math.h>

// ---------------- problem constants (match reference) ----------------
#define GN      50000            // nodes
#define GE      800000           // edges (before self loops)
#define GET     (GE + GN)        // edges incl. self loops
#define FIN     128              // input channels
#define FH      128              // hidden channels (8 heads * 16)
#define NHEADS  8
#define HCH     16
#define FOUT    40
#define NEG_SLOPE 0.2f
#define BN_EPS    1e-5f

typedef __attribute__((ext_vector_type(2))) float v2f;
typedef __attribute__((ext_vector_type(8))) float v8f;
typedef __attribute__((ext_vector_type(4))) int   v4i;

#define AS1 __attribute__((address_space(1)))
#define AS3 __attribute__((address_space(3)))

// 16B global -> LDS copy; async path (ASYNCcnt-tracked) on gfx1250
__device__ __forceinline__ void copy16_g2l(const float* __restrict__ g,
                                           float* __restrict__ l) {
#if defined(__has_builtin) && __has_builtin(__builtin_amdgcn_global_load_async_to_lds_b128)
  __builtin_amdgcn_global_load_async_to_lds_b128(
      (AS1 v4i*)g, (AS3 v4i*)l, /*imm_offset=*/0, /*cpol=*/0);
#else
  *(float4*)l = *(const float4*)g;
#endif
}
__device__ __forceinline__ void g2l_wait() {
#if defined(__has_builtin) && __has_builtin(__builtin_amdgcn_s_wait_asynccnt)
  __builtin_amdgcn_s_wait_asynccnt(0);
#endif
}

// float atomic max via signed/unsigned integer ordering trick
__device__ __forceinline__ void atomic_max_f32(float* addr, float val) {
  if (val >= 0.f) atomicMax((int*)addr, __float_as_int(val));
  else            atomicMin((unsigned int*)addr, __float_as_uint(val));
}

// ---------------------------------------------------------------------
// H[row, col] = sum_k A[row,k] * W[col,k];  K = 128 fixed.
// Block = NWAVES waves; block computes 64 rows x (16*NWAVES) cols.
// W (OUTC x 128, zero-padded to WROWS) and the 64x128 A tile are staged in
// LDS (row stride 132 floats: 16B-aligned, 4-bank stagger => conflict-free
// ds_load_b64 fragment reads). Inner loop: V_WMMA_F32_16X16X4_F32.
//
// fp32 WMMA fragment layouts (CDNA5 ISA 7.12.2):
//   A 16x4: lanes 0-15 = rows (K0/K1 in 2 VGPRs); lanes 16-31 = K2/K3.
//   B 4x16: lanes hold N; low half K0/K1, high half K2/K3.
//   C/D 16x16: VGPR r, lane L -> M = r + 8*(L>=16), N = L%16.
// ---------------------------------------------------------------------
template <int OUTC, int NWAVES>
__global__ void __launch_bounds__(NWAVES * 32) gat_gemm_wmma(
    const float* __restrict__ A, const float* __restrict__ W,
    float* __restrict__ Hout, int nrows) {
  constexpr int K     = 128;
  constexpr int LDK   = K + 4;                    // padded LDS row stride
  constexpr int WROWS = ((OUTC + 15) / 16) * 16;  // zero-padded W rows
  __shared__ float ldsW[WROWS * LDK];
  __shared__ float ldsA[64 * LDK];

  const int tid = threadIdx.x;
  const int m0  = blockIdx.x * 64;

  // ---- stage W: OUTC x 128 (pad rows to WROWS with zeros) ----
  for (int c = tid; c < WROWS * (K / 4); c += NWAVES * 32) {
    const int row = c / (K / 4);
    const int k4  = (c - row * (K / 4)) * 4;
    float* lp = &ldsW[row * LDK + k4];
    if (row < OUTC) copy16_g2l(&W[(size_t)row * K + k4], lp);
    else { lp[0] = 0.f; lp[1] = 0.f; lp[2] = 0.f; lp[3] = 0.f; }
  }
  // ---- stage A tile: 64 x 128 (zero-fill rows past nrows) ----
  for (int c = tid; c < 64 * (K / 4); c += NWAVES * 32) {
    const int row = c / (K / 4);
    const int k4  = (c - row * (K / 4)) * 4;
    float* lp = &ldsA[row * LDK + k4];
    if (m0 + row < nrows) copy16_g2l(&A[(size_t)(m0 + row) * K + k4], lp);
    else { lp[0] = 0.f; lp[1] = 0.f; lp[2] = 0.f; lp[3] = 0.f; }
  }
  g2l_wait();
  __syncthreads();

  const int lane = tid & 31;
  const int wave = tid >> 5;
  const int half = lane >> 4;  // 0: lanes 0-15, 1: lanes 16-31
  const int l16  = lane & 15;
  const int col  = wave * 16 + l16;

  v8f acc[4];
  acc[0] = v8f{}; acc[1] = v8f{}; acc[2] = v8f{}; acc[3] = v8f{};

  const float* wbase = &ldsW[col * LDK + 2 * half];
  const float* abase = &ldsA[l16 * LDK + 2 * half];
#pragma unroll
  for (int kk = 0; kk < K; kk += 4) {
    const v2f b = *(const v2f*)(wbase + kk);
#pragma unroll
    for (int t = 0; t < 4; ++t) {
      const v2f a = *(const v2f*)(abase + t * 16 * LDK + kk);
      acc[t] = __builtin_amdgcn_wmma_f32_16x16x4_f32(
          false, a, false, b, (short)0, acc[t], false, false);
    }
  }

  if (col >= OUTC) return;
#pragma unroll
  for (int t = 0; t < 4; ++t) {
#pragma unroll
    for (int r = 0; r < 8; ++r) {
      const int row = m0 + 16 * t + r + 8 * half;
      if (row < nrows) Hout[(size_t)row * OUTC + col] = acc[t][r];
    }
  }
}

// alpha_src/dst[node,head] = <h[node,head,:], a_src/dst[head,:]>
template <int H, int C>
__global__ void alpha_kernel(const float* __restrict__ Hm,
                             const float* __restrict__ a_src,
                             const float* __restrict__ a_dst,
                             float* __restrict__ as, float* __restrict__ ad,
                             int n) {
  const int i = blockIdx.x * blockDim.x + threadIdx.x;
  if (i >= n * H) return;
  const int node = i / H, hd = i - (i / H) * H;
  const float* hp = Hm + (size_t)node * (H * C) + hd * C;
  const float* s0 = a_src + hd * C;
  const float* d0 = a_dst + hd * C;
  float s = 0.f, d = 0.f;
#pragma unroll
  for (int c = 0; c < C; ++c) { const float v = hp[c]; s += v * s0[c]; d += v * d0[c]; }
  as[i] = s; ad[i] = d;
}

// amax = -inf, denom = 0, out-accumulator = 0
__global__ void init_kernel(float* __restrict__ amax, float* __restrict__ denom,
                            float* __restrict__ out, int nh, int nout) {
  const int i = blockIdx.x * blockDim.x + threadIdx.x;
  if (i < nh)   { amax[i] = -__builtin_inff(); denom[i] = 0.f; }
  if (i < nout) out[i] = 0.f;
}

// pass 1: e = leaky_relu(asrc[src]+adst[dst]); segment max over dst
template <int H>
__global__ void edge_max_kernel(const int* __restrict__ ei,
                                const float* __restrict__ as,
                                const float* __restrict__ ad,
                                float* __restrict__ ew, float* __restrict__ amax,
                                int E, int n) {
  const int i = blockIdx.x * blockDim.x + threadIdx.x;
  if (i >= (E + n) * H) return;
  const int e = i / H, hd = i - e * H;
  int s, d;
  if (e < E) { s = ei[e]; d = ei[E + e]; } else { s = d = e - E; }
  float v = as[s * H + hd] + ad[d * H + hd];
  v = v > 0.f ? v : NEG_SLOPE * v;
  ew[i] = v;
  atomic_max_f32(&amax[d * H + hd], v);
}

// pass 2: w = exp(e - max); segment sum over dst
template <int H>
__global__ void edge_exp_kernel(const int* __restrict__ ei,
                                float* __restrict__ ew,
                                const float* __restrict__ amax,
                                float* __restrict__ denom, int E, int n) {
  const int i = blockIdx.x * blockDim.x + threadIdx.x;
  if (i >= (E + n) * H) return;
  const int e = i / H, hd = i - e * H;
  const int d = (e < E) ? ei[E + e] : e - E;
  const float w = expf(ew[i] - amax[d * H + hd]);
  ew[i] = w;
  atomicAdd(&denom[d * H + hd], w);
}

// pass 3: out[dst] += h[src] * (w / denom[dst]); b128 gathers + fp32 L2 atomics
template <int H, int C>
__global__ void edge_aggr_kernel(const int* __restrict__ ei,
                                 const float* __restrict__ Hm,
                                 const float* __restrict__ ew,
                                 const float* __restrict__ denom,
                                 float* __restrict__ out, int E, int n) {
  const int i = blockIdx.x * blockDim.x + threadIdx.x;
  if (i >= (E + n) * H) return;
  const int e = i / H, hd = i - e * H;
  int s, d;
  if (e < E) { s = ei[e]; d = ei[E + e]; } else { s = d = e - E; }
  const float alpha = ew[i] / (denom[d * H + hd] + 1e-16f);
  const float4* hp4 = (const float4*)(Hm + (size_t)s * (H * C) + hd * C);
  float*        op  = out + (size_t)d * (H * C) + hd * C;
#pragma unroll
  for (int c4 = 0; c4 < C / 4; ++c4) {
    const float4 hv = hp4[c4];
    atomicAdd(&op[c4 * 4 + 0], hv.x * alpha);
    atomicAdd(&op[c4 * 4 + 1], hv.y * alpha);
    atomicAdd(&op[c4 * 4 + 2], hv.z * alpha);
    atomicAdd(&op[c4 * 4 + 3], hv.w * alpha);
  }
}

// x = elu( batchnorm( x + bias ) )
__global__ void bn_elu_kernel(float* __restrict__ x, const float* __restrict__ bias,
                              const float* __restrict__ g, const float* __restrict__ b,
                              const float* __restrict__ m, const float* __restrict__ v,
                              int n, int cdim) {
  const int i = blockIdx.x * blockDim.x + threadIdx.x;
  if (i >= n * cdim) return;
  const int c = i % cdim;
  float val = x[i] + bias[c];
  val = (val - m[c]) * rsqrtf(v[c] + BN_EPS) * g[c] + b[c];
  x[i] = val > 0.f ? val : expm1f(val);
}

__global__ void bias_kernel(float* __restrict__ x, const float* __restrict__ bias,
                            int n, int cdim) {
  const int i = blockIdx.x * blockDim.x + threadIdx.x;
  if (i >= n * cdim) return;
  x[i] += bias[i % cdim];
}

// ---------------------------------------------------------------------
extern "C" void kernel_launch(void* const* d_in, const int* in_sizes, int n_in,
                              void* d_out, int out_size, void* d_ws, size_t ws_size,
                              hipStream_t stream) {
  const float* x    = (const float*)d_in[0];
  const int*   ei   = (const int*)d_in[1];     // [2,E] row-major
  const float* W0   = (const float*)d_in[2];
  const float* as0  = (const float*)d_in[3];
  const float* ad0  = (const float*)d_in[4];
  const float* b0   = (const float*)d_in[5];
  const float* g0   = (const float*)d_in[6];
  const float* bb0  = (const float*)d_in[7];
  const float* mm0  = (const float*)d_in[8];
  const float* vv0  = (const float*)d_in[9];
  const float* W1   = (const float*)d_in[10];
  const float* as1  = (const float*)d_in[11];
  const float* ad1  = (const float*)d_in[12];
  const float* b1   = (const float*)d_in[13];
  const float* g1   = (const float*)d_in[14];
  const float* bb1  = (const float*)d_in[15];
  const float* mm1  = (const float*)d_in[16];
  const float* vv1  = (const float*)d_in[17];
  const float* W2   = (const float*)d_in[18];
  const float* as2  = (const float*)d_in[19];
  const float* ad2  = (const float*)d_in[20];
  const float* b2   = (const float*)d_in[21];
  float* out = (float*)d_out;

  // workspace carve-up (256B aligned regions), ~85 MB total
  size_t off = 0;
  auto walloc = [&](size_t bytes) -> float* {
    void* p = (char*)d_ws + off;
    off += (bytes + 255) & ~(size_t)255;
    return (float*)p;
  };
  float* hbuf = walloc((size_t)GN * FH * 4);        // pre-aggregation features
  float* obuf = walloc((size_t)GN * FH * 4);        // aggregated / next-layer input
  float* asrc = walloc((size_t)GN * NHEADS * 4);
  float* adst = walloc((size_t)GN * NHEADS * 4);
  float* amax = walloc((size_t)GN * NHEADS * 4);
  float* den  = walloc((size_t)GN * NHEADS * 4);
  float* ew   = walloc((size_t)GET * NHEADS * 4);   // per-edge staged e / weight

  const int TPB = 256;
  auto cdiv = [](long long a, long long b) { return (int)((a + b - 1) / b); };
  const int gN64 = cdiv(GN, 64);
  const int gNH  = cdiv((long long)GN * NHEADS, TPB);
  const int gNF  = cdiv((long long)GN * FH, TPB);
  const int gEH  = cdiv((long long)GET * NHEADS, TPB);
  const int gE1  = cdiv(GET, TPB);

  // ---------------- layer 0 ----------------
  gat_gemm_wmma<FH, 8><<<gN64, 256, 0, stream>>>(x, W0, hbuf, GN);
  alpha_kernel<NHEADS, HCH><<<gNH, TPB, 0, stream>>>(hbuf, as0, ad0, asrc, adst, GN);
  init_kernel<<<gNF, TPB, 0, stream>>>(amax, den, obuf, GN * NHEADS, GN * FH);
  edge_max_kernel<NHEADS><<<gEH, TPB, 0, stream>>>(ei, asrc, adst, ew, amax, GE, GN);
  edge_exp_kernel<NHEADS><<<gEH, TPB, 0, stream>>>(ei, ew, amax, den, GE, GN);
  edge_aggr_kernel<NHEADS, HCH><<<gEH, TPB, 0, stream>>>(ei, hbuf, ew, den, obuf, GE, GN);
  bn_elu_kernel<<<gNF, TPB, 0, stream>>>(obuf, b0, g0, bb0, mm0, vv0, GN, FH);

  // ---------------- layer 1 ----------------
  gat_gemm_wmma<FH, 8><<<gN64, 256, 0, stream>>>(obuf, W1, hbuf, GN);
  alpha_kernel<NHEADS, HCH><<<gNH, TPB, 0, stream>>>(hbuf, as1, ad1, asrc, adst, GN);
  init_kernel<<<gNF, TPB, 0, stream>>>(amax, den, obuf, GN * NHEADS, GN * FH);
  edge_max_kernel<NHEADS><<<gEH, TPB, 0, stream>>>(ei, asrc, adst, ew, amax, GE, GN);
  edge_exp_kernel<NHEADS><<<gEH, TPB, 0, stream>>>(ei, ew, amax, den, GE, GN);
  edge_aggr_kernel<NHEADS, HCH><<<gEH, TPB, 0, stream>>>(ei, hbuf, ew, den, obuf, GE, GN);
  bn_elu_kernel<<<gNF, TPB, 0, stream>>>(obuf, b1, g1, bb1, mm1, vv1, GN, FH);

  // ---------------- layer 2 (heads=1, ch=40, mean over 1 head = identity) ----
  gat_gemm_wmma<FOUT, 3><<<gN64, 96, 0, stream>>>(obuf, W2, hbuf, GN);
  alpha_kernel<1, FOUT><<<cdiv(GN, TPB), TPB, 0, stream>>>(hbuf, as2, ad2, asrc, adst, GN);
  init_kernel<<<cdiv((long long)GN * FOUT, TPB), TPB, 0, stream>>>(amax, den, out, GN, GN * FOUT);
  edge_max_kernel<1><<<gE1, TPB, 0, stream>>>(ei, asrc, adst, ew, amax, GE, GN);
  edge_exp_kernel<1><<<gE1, TPB, 0, stream>>>(ei, ew, amax, den, GE, GN);
  edge_aggr_kernel<1, FOUT><<<gE1, TPB, 0, stream>>>(ei, hbuf, ew, den, out, GE, GN);
  bias_kernel<<<cdiv((long long)GN * FOUT, TPB), TPB, 0, stream>>>(out, b2, GN, FOUT);
}